// TiSASRec_3384434229849
// MI455X (gfx1250) — compile-verified
//
#include <hip/hip_runtime.h>
#include <hip/hip_bf16.h>

typedef __attribute__((ext_vector_type(2))) float v2f;
typedef __attribute__((ext_vector_type(8))) float v8f;

namespace {
constexpr int B  = 8;
constexpr int L  = 200;
constexpr int H  = 128;
constexpr int NH = 4;
constexpr int DH = 32;
constexpr int R  = B * L;              // 1600 rows, exactly 100 16-row tiles
constexpr float SCALE = 0.1767766952966369f;   // 1/sqrt(32)
constexpr float NEGV  = -4294967295.0f;        // -2^32 + 1
}

static __device__ __forceinline__ v8f wmma_f32(v2f a, v2f b, v8f c) {
  return __builtin_amdgcn_wmma_f32_16x16x4_f32(false, a, false, b, (short)0, c,
                                               false, false);
}

// ---------------------------------------------------------------------------
// Kernel 1: fused projections.
//   Q  = queries @ Qw^T + Qb
//   Kp = keys    @ Kw^T + Kb + abs_pos_K     (absorbs Q·pK^T bias term)
//   Vp = keys    @ Vw^T + Vb + abs_pos_V     (absorbs P·pV output term)
// One wave per 16x16 tile; K=128 reduced in 32 steps of 4 -> 96 WMMAs/wave.
// ---------------------------------------------------------------------------
__global__ __launch_bounds__(32) void tisas_proj_kernel(
    const float* __restrict__ queries, const float* __restrict__ keys,
    const float* __restrict__ Qw, const float* __restrict__ Qb,
    const float* __restrict__ Kw, const float* __restrict__ Kb,
    const float* __restrict__ Vw, const float* __restrict__ Vb,
    const float* __restrict__ posK, const float* __restrict__ posV,
    float* __restrict__ Qo, float* __restrict__ Kpo, float* __restrict__ Vpo) {
  const int lane = threadIdx.x & 31;
  const int r0 = blockIdx.x * 16;     // row tile  (R/16 = 100 exact)
  const int c0 = blockIdx.y * 16;     // col tile  (H/16 = 8 exact)
  const int mr = lane & 15;           // M (or N) index inside tile
  const int kb = (lane >> 4) * 2;     // K sub-slot {0,2} per ISA A/B layout

  const float* qrow = queries + (size_t)(r0 + mr) * H;
  const float* krow = keys    + (size_t)(r0 + mr) * H;
  const float* wq   = Qw + (size_t)(c0 + mr) * H;   // B[k][n] = W[c0+n][k]
  const float* wk   = Kw + (size_t)(c0 + mr) * H;
  const float* wv   = Vw + (size_t)(c0 + mr) * H;

  v8f accQ = {}; v8f accK = {}; v8f accV = {};
  for (int kk = 0; kk < H; kk += 4) {
    v2f aq = { qrow[kk + kb], qrow[kk + kb + 1] };
    v2f ak = { krow[kk + kb], krow[kk + kb + 1] };
    v2f bq = { wq[kk + kb],   wq[kk + kb + 1] };
    v2f bk = { wk[kk + kb],   wk[kk + kb + 1] };
    v2f bv = { wv[kk + kb],   wv[kk + kb + 1] };
    accQ = wmma_f32(aq, bq, accQ);
    accK = wmma_f32(ak, bk, accK);
    accV = wmma_f32(ak, bv, accV);
  }

  const int c = c0 + mr;              // D: n = lane&15, m = i + 8*(lane>>4)
  const float qb = Qb[c], kbia = Kb[c], vb = Vb[c];
#pragma unroll
  for (int i = 0; i < 8; ++i) {
    const int row = r0 + i + 8 * (lane >> 4);
    const size_t idx = (size_t)row * H + c;
    Qo[idx]  = accQ[i] + qb;
    Kpo[idx] = accK[i] + kbia + posK[idx];
    Vpo[idx] = accV[i] + vb   + posV[idx];
  }
}

// ---------------------------------------------------------------------------
// Kernel 2: attention for one (b, head, 16-query tile).
//  S  = Q @ Kp^T              (WMMA, 8 waves split 13 key tiles)
//  S += batched GEMV tK·q     (streaming, 128B-coalesced float4 + shfl reduce)
//  softmax with causal + time mask (NEG sentinel reproduces uniform rows)
//  outT = batched GEMV P·tV   (streaming, 32-lane coalesced over d)
//  out  = P @ Vp + outT       (WMMA, waves 0..1)
// ---------------------------------------------------------------------------
__global__ __launch_bounds__(256) void tisas_attn_kernel(
    const float* __restrict__ Qbuf, const float* __restrict__ Kp,
    const float* __restrict__ Vp, const float* __restrict__ tK,
    const float* __restrict__ tV, const int* __restrict__ time_mask,
    float* __restrict__ out) {
  __shared__ float Qs[16][36];    // stride 36: 16B-aligned rows, conflict-free
  __shared__ float S[16][210];    // stride 210: conflict-free column reads
  __shared__ float outT[16][32];

  const int b  = blockIdx.z;
  const int h  = blockIdx.y;
  const int l0 = blockIdx.x * 16;
  const int tid  = threadIdx.x;
  const int wave = tid >> 5;
  const int lane = tid & 31;
  const int mr = lane & 15;
  const int kb = (lane >> 4) * 2;

  // ---- stage Q tile (rows l0..l0+15, this head's 32 dims) into LDS ----
  for (int i = tid; i < 16 * DH; i += 256) {
    const int l = i >> 5, d = i & 31, lr = l0 + l;
    Qs[l][d] = (lr < L) ? Qbuf[(size_t)(b * L + lr) * H + h * DH + d] : 0.f;
  }
  __syncthreads();

  // A fragments of the Q tile (shared by every key tile this wave handles)
  v2f afr[8];
#pragma unroll
  for (int ks = 0; ks < 8; ++ks) {
    const int kk = ks * 4 + kb;
    afr[ks] = { Qs[mr][kk], Qs[mr][kk + 1] };
  }

  // ---- S = Q @ Kp^T : 13 key tiles over 8 waves ----
  for (int mt = wave; mt < 13; mt += 8) {
    const int m0 = mt * 16;
    const int m  = m0 + mr;
    const bool mv = (m < L);
    const float* kpb = Kp + (size_t)(b * L + (mv ? m : 0)) * H + h * DH;
    v8f acc = {};
#pragma unroll
    for (int ks = 0; ks < 8; ++ks) {
      const int kk = ks * 4 + kb;
      v2f bf = { mv ? kpb[kk] : 0.f, mv ? kpb[kk + 1] : 0.f };
      acc = wmma_f32(afr[ks], bf, acc);
    }
#pragma unroll
    for (int i = 0; i < 8; ++i)
      S[i + 8 * (lane >> 4)][m0 + mr] = acc[i];
  }
  __syncthreads();

  // ---- relative-time key bias: S[l][m] += tK[b,l,m,h,:] . Q[l,:] ----
  // 8 lanes cooperate per m (float4 each => 128B contiguous per group),
  // 4 m's per wave per step, shfl_xor tree reduction within the 8-lane group.
  {
    const int g = lane >> 3, sub = lane & 7;
    for (int l = 0; l < 16; ++l) {
      const int lr = l0 + l;
      if (lr >= L) break;
      const float4 q = *(const float4*)&Qs[l][sub * 4];
      const float* tkb =
          tK + (size_t)(b * L + lr) * L * H + h * DH + sub * 4;
      for (int m = wave * 4 + g; m < L; m += 32) {
        const float4 v = *(const float4*)(tkb + (size_t)m * H);
        float p = v.x * q.x + v.y * q.y + v.z * q.z + v.w * q.w;
        p += __shfl_xor(p, 1, 32);
        p += __shfl_xor(p, 2, 32);
        p += __shfl_xor(p, 4, 32);
        if (sub == 0) S[l][m] += p;
      }
    }
  }
  __syncthreads();

  // ---- scale + mask + softmax (one wave per row, rows wave, wave+8) ----
  for (int l = wave; l < 16; l += 8) {
    const int lr = l0 + l;
    const bool rowmask = time_mask[b * L + (lr < L ? lr : L - 1)] != 0;
    float mx = -3.4e38f;
    for (int m = lane; m < L; m += 32) {
      float v = S[l][m] * SCALE;
      if (rowmask || m > lr) v = NEGV;   // mask AFTER scaling, like reference
      S[l][m] = v;
      mx = fmaxf(mx, v);
    }
#pragma unroll
    for (int o = 16; o > 0; o >>= 1) mx = fmaxf(mx, __shfl_xor(mx, o, 32));
    float sum = 0.f;
    for (int m = lane; m < L; m += 32) {
      const float e = __expf(S[l][m] - mx);
      S[l][m] = e;
      sum += e;
    }
#pragma unroll
    for (int o = 16; o > 0; o >>= 1) sum += __shfl_xor(sum, o, 32);
    const float inv = 1.f / sum;
    for (int m = lane; m < 210; m += 32)
      S[l][m] = (m < L) ? S[l][m] * inv : 0.f;   // zero-pad P for WMMA K-tail
  }
  __syncthreads();

  // ---- relative-time value term: outT[l][d] = sum_m P[l][m]*tV[b,l,m,h,d]
  // 32 consecutive lanes cover d=0..31 => 128B coalesced per m step.
  for (int p = tid; p < 512; p += 256) {
    const int l = p >> 5, d = p & 31, lr = l0 + l;
    float acc = 0.f;
    if (lr < L) {
      const float* tvb = tV + (size_t)(b * L + lr) * L * H + h * DH + d;
      for (int m = 0; m < L; ++m)
        acc += S[l][m] * tvb[(size_t)m * H];
    }
    outT[l][d] = acc;
  }
  __syncthreads();

  // ---- out = P @ Vp + outT : waves 0,1 own the two 16-wide d tiles ----
  if (wave < 2) {
    const int n0 = wave * 16;
    v8f acc = {};
    for (int kt = 0; kt < 52; ++kt) {           // K=208 (P zero-padded)
      const int m = kt * 4 + kb;
      v2f af = { S[mr][m], S[mr][m + 1] };
      v2f bf = { 0.f, 0.f };
      const float* vb = Vp + (size_t)(b * L + m) * H + h * DH + n0 + mr;
      if (m < L)     bf.x = vb[0];
      if (m + 1 < L) bf.y = vb[H];
      acc = wmma_f32(af, bf, acc);
    }
#pragma unroll
    for (int i = 0; i < 8; ++i) {
      const int mrow = i + 8 * (lane >> 4);
      const int lr = l0 + mrow;
      if (lr < L)
        out[(size_t)(b * L + lr) * H + h * DH + n0 + mr] =
            acc[i] + outT[mrow][n0 + mr];
    }
  }
}

extern "C" void kernel_launch(void* const* d_in, const int* in_sizes, int n_in,
                              void* d_out, int out_size, void* d_ws,
                              size_t ws_size, hipStream_t stream) {
  (void)in_sizes; (void)n_in; (void)out_size; (void)ws_size;
  const float* queries  = (const float*)d_in[0];
  const float* keys     = (const float*)d_in[1];
  const int*   tmask    = (const int*)d_in[2];    // bool mask -> int per harness rule
  /* d_in[3] attn_mask: deterministic triu(k=1); computed as (m > l) in-kernel */
  const float* tK       = (const float*)d_in[4];
  const float* tV       = (const float*)d_in[5];
  const float* posK     = (const float*)d_in[6];
  const float* posV     = (const float*)d_in[7];
  const float* Qw       = (const float*)d_in[8];
  const float* Qb       = (const float*)d_in[9];
  const float* Kw       = (const float*)d_in[10];
  const float* Kb       = (const float*)d_in[11];
  const float* Vw       = (const float*)d_in[12];
  const float* Vb       = (const float*)d_in[13];
  float* out = (float*)d_out;

  // workspace: Q, Kp=K+posK, Vp=V+posV, each [B*L, H] f32 (2.4 MB total)
  float* Qbuf = (float*)d_ws;
  float* Kp   = Qbuf + (size_t)R * H;
  float* Vp   = Kp   + (size_t)R * H;

  tisas_proj_kernel<<<dim3(R / 16, H / 16), 32, 0, stream>>>(
      queries, keys, Qw, Qb, Kw, Kb, Vw, Vb, posK, posV, Qbuf, Kp, Vp);

  tisas_attn_kernel<<<dim3((L + 15) / 16, NH, B), 256, 0, stream>>>(
      Qbuf, Kp, Vp, tK, tV, tmask, out);
}